// color_entropy_loss_81037442941133
// MI455X (gfx1250) — compile-verified
//
#include <hip/hip_runtime.h>
#include <cstdint>

// ---------------------------------------------------------------------------
// Problem constants (fixed by the reference: x is (32, 3, 512, 512) fp32)
// ---------------------------------------------------------------------------
#define BATCH          32
#define HW             (512 * 512)          // pixels per channel plane
#define NBINS          256
#define NUM_PIXEL      (HW + NBINS)         // 262400, reference's num_pixel
#define THREADS        256                  // 8 wave32 per block
#define WAVES          8
#define BLOCKS_PER_B   32                   // 1024 workgroups total
#define F4_PER_PLANE   (HW / 4)             // 65536 float4 per (b,c) plane
#define F4_PER_BLOCK   (F4_PER_PLANE / BLOCKS_PER_B)  // 2048
#define TILE4          512                  // float4 per channel per LDS tile
#define NTILES         (F4_PER_BLOCK / TILE4)         // 4

// ---------------------------------------------------------------------------
// CDNA5 async global->LDS copy path (ASYNCcnt). Inline asm: portable across
// ROCm-7.2 clang-22 and amdgpu-toolchain clang-23 (builtin arity differs).
// Input is a single-touch 100.7 MB stream -> non-temporal load hint.
// ---------------------------------------------------------------------------
__device__ __forceinline__ unsigned lds_off_u32(const void* p) {
  // Flat pointers into the LDS aperture carry the LDS byte offset in [31:0]
  // (CDNA5 ISA aperture mapping).
  return (unsigned)(unsigned long long)p;
}

__device__ __forceinline__ void async_copy_b128_nt(unsigned lds_off, const void* gptr) {
  // Per active lane: 16B global -> LDS, no VGPR data staging, ASYNCcnt++.
  asm volatile("global_load_async_to_lds_b128 %0, %1, off th:TH_LOAD_NT"
               :: "v"(lds_off), "v"(gptr)
               : "memory");
}

#if defined(__has_builtin)
#if __has_builtin(__builtin_amdgcn_s_wait_asynccnt)
#define WAIT_ASYNC(n) __builtin_amdgcn_s_wait_asynccnt(n)
#endif
#endif
#ifndef WAIT_ASYNC
#define WAIT_ASYNC(n) asm volatile("s_wait_asynccnt %0" :: "i"(n) : "memory")
#endif

// ---------------------------------------------------------------------------
// Kernel 0: zero the per-batch global histograms (d_ws is poisoned 0xAA).
// ---------------------------------------------------------------------------
__global__ void ce_zero_kernel(unsigned* __restrict__ g_hist) {
  int i = blockIdx.x * blockDim.x + threadIdx.x;
  if (i < BATCH * NBINS) g_hist[i] = 0u;
}

// ---------------------------------------------------------------------------
// Kernel 1: histogram build.
//   grid = BATCH * BLOCKS_PER_B blocks, 256 threads each.
//   Double-buffered async staging: 2 x (3 channels x 512 float4) = 48 KB LDS,
//   plus 8 per-wave 256-bin histograms (8 KB) to minimize ds_add contention.
//   Per wave: 6 async b128 ops in flight per tile (512 B each) x 2 buffers.
// ---------------------------------------------------------------------------
__global__ __launch_bounds__(THREADS)
void ce_hist_kernel(const float* __restrict__ x, unsigned* __restrict__ g_hist) {
  __shared__ unsigned hist[WAVES * NBINS];        // 8 KB
  __shared__ float4   stage[2][3 * TILE4];        // 48 KB

  const int tid  = threadIdx.x;
  const int wave = tid >> 5;                      // wave32
  const int b    = blockIdx.x / BLOCKS_PER_B;
  const int blk  = blockIdx.x % BLOCKS_PER_B;

  for (int i = tid; i < WAVES * NBINS; i += THREADS) hist[i] = 0u;

  const float4* x4 = (const float4*)x;
  const int base4  = blk * F4_PER_BLOCK;          // float4 offset within plane

  // Issue all async loads for one tile: 3 channels x 2 rounds x 256 threads
  // -> 6 ASYNCcnt increments per wave, 512 B moved per wave per instruction.
  auto issue = [&](int t, int buf) {
    const int tb = base4 + t * TILE4;
#pragma unroll
    for (int c = 0; c < 3; ++c) {
      const float4* src = x4 + (size_t)(b * 3 + c) * F4_PER_PLANE + tb;
#pragma unroll
      for (int r = 0; r < 2; ++r) {
        const int i = tid + r * THREADS;
        async_copy_b128_nt(lds_off_u32(&stage[buf][c * TILE4 + i]), src + i);
      }
    }
  };

  // Bin 8 pixels (2 float4 per channel) from LDS; exact reference arithmetic:
  // temp = (x0*255 + x1*255 + x2*255) / 3, idx = clip(trunc(temp), 0, 255).
  auto process = [&](int buf) {
#pragma unroll
    for (int r = 0; r < 2; ++r) {
      const int i = tid * 2 + r;
      const float4 a = stage[buf][0 * TILE4 + i];
      const float4 g = stage[buf][1 * TILE4 + i];
      const float4 u = stage[buf][2 * TILE4 + i];
      const float av[4] = {a.x, a.y, a.z, a.w};
      const float gv[4] = {g.x, g.y, g.z, g.w};
      const float uv[4] = {u.x, u.y, u.z, u.w};
#pragma unroll
      for (int k = 0; k < 4; ++k) {
        float s = __fadd_rn(__fadd_rn(__fmul_rn(av[k], 255.0f),
                                      __fmul_rn(gv[k], 255.0f)),
                            __fmul_rn(uv[k], 255.0f));
        int bin = (int)__fdiv_rn(s, 3.0f);        // trunc toward zero, s >= 0
        bin = bin < 0 ? 0 : (bin > 255 ? 255 : bin);
        atomicAdd(&hist[wave * NBINS + bin], 1u); // ds_add_u32
      }
    }
  };

  issue(0, 0);
  for (int t = 0; t < NTILES; ++t) {
    const int buf = t & 1;
    if (t + 1 < NTILES) {
      issue(t + 1, buf ^ 1);   // prefetch next tile into the other buffer
      WAIT_ASYNC(6);           // own wave's tile-t copies (6 older ops) done
    } else {
      WAIT_ASYNC(0);
    }
    __syncthreads();           // every wave waited -> tile t fully in LDS
    process(buf);
    __syncthreads();           // all reads of buf done before it is reissued
  }

  // Reduce 8 per-wave histograms, flush with deterministic integer atomics.
  unsigned sum = 0u;
#pragma unroll
  for (int w = 0; w < WAVES; ++w) sum += hist[w * NBINS + tid];
  atomicAdd(&g_hist[b * NBINS + tid], sum);       // global_atomic_add_u32
}

// ---------------------------------------------------------------------------
// Kernel 2: entropy. 1 block, thread t owns bin t across all 32 batches.
// Reproduces the reference quirks: counts[:,0] = H*W, then +1 smoothing.
// Fixed-order LDS tree reduction -> deterministic output.
// ---------------------------------------------------------------------------
__global__ __launch_bounds__(THREADS)
void ce_entropy_kernel(const unsigned* __restrict__ g_hist, float* __restrict__ out) {
  __shared__ float red[THREADS];
  const int t = threadIdx.x;

  float acc = 0.0f;
  const float inv_np = 1.0f / (float)NUM_PIXEL;
  for (int b = 0; b < BATCH; ++b) {
    float cnt = (t == 0) ? (float)HW : (float)g_hist[b * NBINS + t];
    float p = (cnt + 1.0f) * inv_np;
    acc += p * logf(p);
  }
  red[t] = acc;
  __syncthreads();
#pragma unroll
  for (int s = THREADS / 2; s > 0; s >>= 1) {
    if (t < s) red[t] += red[t + s];
    __syncthreads();
  }
  if (t == 0) out[0] = red[0] * (1.0f / (float)BATCH);
}

// ---------------------------------------------------------------------------
// Launch: zero ws histogram -> build histograms -> entropy.
// ---------------------------------------------------------------------------
extern "C" void kernel_launch(void* const* d_in, const int* in_sizes, int n_in,
                              void* d_out, int out_size, void* d_ws, size_t ws_size,
                              hipStream_t stream) {
  (void)in_sizes; (void)n_in; (void)out_size; (void)ws_size;
  const float* x      = (const float*)d_in[0];
  float*       out    = (float*)d_out;
  unsigned*    g_hist = (unsigned*)d_ws;          // 32*256*4 = 32 KB scratch

  ce_zero_kernel<<<(BATCH * NBINS + THREADS - 1) / THREADS, THREADS, 0, stream>>>(g_hist);
  ce_hist_kernel<<<BATCH * BLOCKS_PER_B, THREADS, 0, stream>>>(x, g_hist);
  ce_entropy_kernel<<<1, THREADS, 0, stream>>>(g_hist, out);
}